// DifferentiablePDESolver_73237782331835
// MI455X (gfx1250) — compile-verified
//
#include <hip/hip_runtime.h>
#include <cstdint>

// Reaction-diffusion PDE, 32 explicit Euler steps, (2,1,128,128,128) f32.
// Bandwidth-bound stencil: working set (67MB) fits MI455X's 192MB L2 -> steps
// 1..31 run at L2 bandwidth. CDNA5 path: global_load_async_to_lds_b32 staging
// (ASYNCcnt) with a depth-1 software-pipelined z-march over 4 LDS plane buffers.

#define NXD 128
constexpr int TX       = 32;              // tile x (one wave32 row)
constexpr int TY       = 8;               // tile y
constexpr int ZSLAB    = 32;              // planes per block
constexpr int NTHREADS = TX * TY;         // 256 threads = 8 wave32
constexpr int PLANE    = (TY + 2) * (TX + 2);   // 340 halo-tile elements
constexpr int NBUF     = 4;               // rotating LDS plane buffers

__device__ __forceinline__ void async_wait_all() {
#if __has_builtin(__builtin_amdgcn_s_wait_asynccnt)
  __builtin_amdgcn_s_wait_asynccnt(0);
#else
  asm volatile("s_wait_asynccnt 0" ::: "memory");
#endif
}

// GVS-mode async copy: copies 4 bytes from global address sbase plus mem_byte_off to LDS address lds_byte_off   (tracked by ASYNCcnt)
__device__ __forceinline__ void async_load_b32(uint32_t lds_byte_off,
                                               uint32_t mem_byte_off,
                                               const float* sbase) {
  asm volatile("global_load_async_to_lds_b32 %0, %1, %2 offset:0"
               :: "v"(lds_byte_off), "v"(mem_byte_off), "s"(sbase)
               : "memory");
}

__device__ __forceinline__ int clampi(int v, int lo, int hi) {
  return v < lo ? lo : (v > hi ? hi : v);
}

__global__ __launch_bounds__(NTHREADS) void pde_step_kernel(
    const float* __restrict__ u, const float* __restrict__ Dmap,
    const float* __restrict__ rmap, const int* __restrict__ delta_t,
    float* __restrict__ out, int step) {
  __shared__ float sm[NBUF][TY + 2][TX + 2];

  const int b  = blockIdx.z >> 2;            // batch
  const int zs = (blockIdx.z & 3) * ZSLAB;   // z-slab start
  const int x0 = blockIdx.x * TX;
  const int y0 = blockIdx.y * TY;
  const int tx = threadIdx.x, ty = threadIdx.y;
  const int tid = ty * TX + tx;

  const size_t V = (size_t)NXD * NXD * NXD;
  const float* ub = u    + (size_t)b * V;
  const float* Db = Dmap + (size_t)b * V;
  const float* rb = rmap + (size_t)b * V;
  float*       ob = out  + (size_t)b * V;
  const float active = (step < delta_t[b]) ? 1.0f : 0.0f;

  // ---- Loop-invariant halo-load geometry (each lane fills <=2 plane slots) ----
  const uint32_t smbase = (uint32_t)(uintptr_t)&sm[0][0][0];  // low 32 bits of the generic pointer are the LDS address
  auto xy_byte_off = [&](int i) -> uint32_t {                 // clamped (y,x) byte offset in a plane
    const int ly = i / (TX + 2);
    const int lx = i - ly * (TX + 2);
    const int gy = clampi(y0 + ly - 1, 0, NXD - 1);
    const int gx = clampi(x0 + lx - 1, 0, NXD - 1);
    return ((uint32_t)gy * NXD + (uint32_t)gx) * 4u;
  };
  const uint32_t xy0 = xy_byte_off(tid);
  const uint32_t lo0 = (uint32_t)tid * 4u;
  const bool     has2 = tid < (PLANE - NTHREADS);             // 84 lanes carry a 2nd slot
  const uint32_t xy1 = xy_byte_off(has2 ? tid + NTHREADS : tid);
  const uint32_t lo1 = (uint32_t)(tid + NTHREADS) * 4u;

  // Async-stage plane zp (z clamped; x/y pre-clamped) into LDS buffer bf.
  auto issue_plane = [&](int zp, int bf) {
    const int zc = clampi(zp, 0, NXD - 1);
    const uint32_t zb = (uint32_t)zc * (NXD * NXD * 4u);
    const uint32_t lb = smbase + (uint32_t)bf * (PLANE * 4u);
    async_load_b32(lb + lo0, zb + xy0, ub);
    if (has2) async_load_b32(lb + lo1, zb + xy1, ub);
  };
  // buf(k) = k & 3 for local plane index k = z - zs  (k = -1 -> buf 3)

  // Prologue: planes k=-1,0 -> grab u_prev -> start k=1
  issue_plane(zs - 1, 3);
  issue_plane(zs,     0);
  async_wait_all();
  __syncthreads();
  float u_prev = sm[3][ty + 1][tx + 1];
  issue_plane(zs + 1, 1);

  const int gx = x0 + tx, gy = y0 + ty;
  const size_t row = (size_t)gy * NXD + gx;

  for (int k = 0; k < ZSLAB; k += NBUF) {
#pragma unroll
    for (int j = 0; j < NBUF; ++j) {       // buffer indices are compile-time: j,(j+1)&3,(j+2)&3
      const int kk = k + j;
      const int z  = zs + kk;

      async_wait_all();                    // plane k+1 landed in LDS
      __syncthreads();                     // visible to all waves; buf(k+2) fully consumed
      if (kk + 2 <= ZSLAB)                 // stream plane k+2 while computing plane k
        issue_plane(z + 2, (j + 2) & 3);

      const float uc  = sm[j][ty + 1][tx + 1];
      const float uxm = sm[j][ty + 1][tx    ];
      const float uxp = sm[j][ty + 1][tx + 2];
      const float uym = sm[j][ty    ][tx + 1];
      const float uyp = sm[j][ty + 2][tx + 1];
      const float uzp = sm[(j + 1) & 3][ty + 1][tx + 1];

      float lap = -6.0f * uc;              // zero-padded 7-point Laplacian
      lap += (gx > 0       ? uxm    : 0.0f);
      lap += (gx < NXD - 1 ? uxp    : 0.0f);
      lap += (gy > 0       ? uym    : 0.0f);
      lap += (gy < NXD - 1 ? uyp    : 0.0f);
      lap += (z  > 0       ? u_prev : 0.0f);
      lap += (z  < NXD - 1 ? uzp    : 0.0f);

      const size_t idx = (size_t)z * (NXD * NXD) + row;
      const float du = Db[idx] * lap + rb[idx] * uc * (1.0f - uc);
      float r = fmaf(active, du, uc);      // DT = 1.0
      r = fminf(fmaxf(r, 0.0f), 1.0f);
      ob[idx] = r;
      u_prev = uc;
    }
  }
}

extern "C" void kernel_launch(void* const* d_in, const int* in_sizes, int n_in,
                              void* d_out, int out_size, void* d_ws, size_t ws_size,
                              hipStream_t stream) {
  const float* u0 = (const float*)d_in[0];
  const float* Dm = (const float*)d_in[1];
  const float* rm = (const float*)d_in[2];
  const int*   dt = (const int*)d_in[3];
  float* out = (float*)d_out;
  float* tmp = (float*)d_ws;   // needs 2*128^3*4 = 16.8 MB of scratch

  const dim3 grid(NXD / TX, NXD / TY, 2 * (NXD / ZSLAB));  // (4,16,8) = 512 blocks
  const dim3 block(TX, TY, 1);

  // 32 sequential steps, ping-pong so step 31 (odd) writes d_out.
  for (int s = 0; s < 32; ++s) {
    const float* src = (s == 0) ? u0 : ((s & 1) ? tmp : out);
    float* dst = (s & 1) ? out : tmp;
    pde_step_kernel<<<grid, block, 0, stream>>>(src, Dm, rm, dt, dst, s);
  }
}